// YourGCN_69011534512459
// MI455X (gfx1250) — compile-verified
//
#include <hip/hip_runtime.h>
#include <hip/hip_bf16.h>
#include <math.h>

typedef float v2f __attribute__((ext_vector_type(2)));
typedef float v8f __attribute__((ext_vector_type(8)));

#define NEG_SLOPE 0.2f

// ---------------------------------------------------------------------------
// monotonic float<->uint encoding for atomicMax-based segment max
// ---------------------------------------------------------------------------
__device__ __forceinline__ unsigned f2mono(float f) {
  unsigned u = __float_as_uint(f);
  return (u & 0x80000000u) ? ~u : (u | 0x80000000u);
}
__device__ __forceinline__ float mono2f(unsigned u) {
  return (u & 0x80000000u) ? __uint_as_float(u ^ 0x80000000u)
                           : __uint_as_float(~u);
}
#define MONO_NEG_INF 0x007FFFFFu  // f2mono(-inf)

// ---------------------------------------------------------------------------
// Tiled fp32 GEMM using V_WMMA_F32_16X16X4_F32.
// Each block: WAVES waves, one 16-row M tile staged in LDS (float4 fill),
// wave w owns N-tile [16w,16w+16). A layout (ISA 7.12.2, 32-bit A 16x4):
// lane = M row, VGPR0/1 = K pair, lane-halves split K {0,1} vs {2,3}.
// B mirrored: lane = N. Full-tile stores are branchless with immediate
// offsets; ragged last tile takes the guarded path.
// ---------------------------------------------------------------------------
template <int K, int N, int WAVES>
__global__ void gat_gemm_wmma(const float* __restrict__ X,
                              const float* __restrict__ W,
                              float* __restrict__ H, int n_nodes) {
  constexpr int LDK = K + 4;  // pad: (K+4)%64 = 4 banks -> conflict-free,
                              // and (K+4)*4B stays 16-B aligned per row
  __shared__ alignas(16) float As[16 * LDK];
  const int row0 = blockIdx.x * 16;
  const int tid = threadIdx.x;

  // stage A tile: float4 per thread-iteration (global_load_b128/ds_store_b128)
  for (int i = tid; i < 16 * (K / 4); i += WAVES * 32) {
    int r = i / (K / 4), c4 = i % (K / 4);
    float4 v = make_float4(0.f, 0.f, 0.f, 0.f);
    if (row0 + r < n_nodes)
      v = ((const float4*)X)[(size_t)(row0 + r) * (K / 4) + c4];
    *(float4*)&As[r * LDK + 4 * c4] = v;
  }
  __syncthreads();

  const int lane = tid & 31;
  const int half = lane >> 4;
  const int l16 = lane & 15;
  const int colbase = (tid >> 5) * 16;

  v8f acc = {};
#pragma unroll 4
  for (int k0 = 0; k0 < K; k0 += 4) {
    const float* ap = &As[l16 * LDK + k0 + 2 * half];
    v2f a;
    a.x = ap[0];
    a.y = ap[1];
    const float* bp = &W[(size_t)(k0 + 2 * half) * N + colbase + l16];
    v2f b;
    b.x = bp[0];
    b.y = bp[N];
    acc = __builtin_amdgcn_wmma_f32_16x16x4_f32(false, a, false, b, (short)0,
                                                acc, false, false);
  }

  // D tile store: lane (l16,half) holds rows mbase..mbase+7, col colbase+l16
  const int mbase = 8 * half;
  float* hp = H + (size_t)(row0 + mbase) * N + colbase + l16;
  if (row0 + 16 <= n_nodes) {
#pragma unroll
    for (int i = 0; i < 8; ++i) hp[i * N] = acc[i];  // immediate offsets
  } else {
#pragma unroll
    for (int i = 0; i < 8; ++i)
      if (row0 + mbase + i < n_nodes) hp[i * N] = acc[i];
  }
}

// ---------------------------------------------------------------------------
// Per-node attention coefficients a_s = h.att_src, a_d = h.att_dst
// plus init of segment-max / segment-sum accumulators. One wave per node.
// ---------------------------------------------------------------------------
template <int C>
__global__ void node_prep(const float* __restrict__ H,
                          const float* __restrict__ att_s,
                          const float* __restrict__ att_d, float* a_s,
                          float* a_d, unsigned* m_enc, float* den,
                          int n_nodes) {
  int node = (int)((blockIdx.x * (size_t)blockDim.x + threadIdx.x) >> 5);
  int lane = threadIdx.x & 31;
  if (node >= n_nodes) return;
  float ps = 0.f, pd = 0.f;
#pragma unroll
  for (int j = 0; j < C / 32; ++j) {
    int ch = lane + 32 * j;
    float h = H[(size_t)node * C + ch];
    ps += h * att_s[ch];
    pd += h * att_d[ch];
  }
#pragma unroll
  for (int off = 16; off > 0; off >>= 1) {
    ps += __shfl_xor(ps, off, 32);
    pd += __shfl_xor(pd, off, 32);
  }
  if (lane == 0) {
    a_s[node] = ps;
    a_d[node] = pd;
    m_enc[node] = MONO_NEG_INF;
    den[node] = 0.f;
  }
}

// out[n][c] = bias[c]
template <int C>
__global__ void init_out(float* __restrict__ out, const float* __restrict__ b,
                         int n_nodes) {
  size_t i = (size_t)blockIdx.x * blockDim.x + threadIdx.x;
  if (i < (size_t)n_nodes * C) out[i] = b[i % C];
}

__device__ __forceinline__ void edge_sd(const long long* __restrict__ ei,
                                        int e, int nE, int& s, int& d) {
  if (e < nE) {
    s = (int)ei[e];
    d = (int)ei[(size_t)nE + e];
  } else {
    s = d = e - nE;  // self loops appended
  }
}

// pass A: logits + segment max (atomicMax on monotonic-encoded float)
__global__ void edge_logits(const long long* __restrict__ ei,
                            const float* __restrict__ a_s,
                            const float* __restrict__ a_d,
                            float* __restrict__ logit, unsigned* m_enc, int nE,
                            int nN) {
  int e = blockIdx.x * blockDim.x + threadIdx.x;
  if (e >= nE + nN) return;
  int s, d;
  edge_sd(ei, e, nE, s, d);
  float z = a_s[s] + a_d[d];
  z = z > 0.f ? z : NEG_SLOPE * z;
  logit[e] = z;
  atomicMax(&m_enc[d], f2mono(z));
}

// pass B: e = exp(logit - max[dst]); segment sum
__global__ void edge_exp(const long long* __restrict__ ei,
                         float* __restrict__ logit,
                         const unsigned* __restrict__ m_enc, float* den,
                         int nE, int nN) {
  int e = blockIdx.x * blockDim.x + threadIdx.x;
  if (e >= nE + nN) return;
  int s, d;
  edge_sd(ei, e, nE, s, d);
  float ex = expf(logit[e] - mono2f(m_enc[d]));
  logit[e] = ex;  // in place (each thread owns its slot)
  atomicAdd(&den[d], ex);
}

// pass C: out[dst] += alpha * h[src] ; one wave per edge, C/32 ch per lane
template <int C>
__global__ void edge_aggregate(const long long* __restrict__ ei,
                               const float* __restrict__ ew,
                               const float* __restrict__ den,
                               const float* __restrict__ H,
                               float* __restrict__ out, int nE, int nN) {
  int e = (int)((blockIdx.x * (size_t)blockDim.x + threadIdx.x) >> 5);
  int lane = threadIdx.x & 31;
  if (e >= nE + nN) return;
  int s, d;
  edge_sd(ei, e, nE, s, d);
  float alpha = ew[e] / den[d];
#pragma unroll
  for (int j = 0; j < C / 32; ++j) {
    int ch = lane + 32 * j;
    atomicAdd(&out[(size_t)d * C + ch], H[(size_t)s * C + ch] * alpha);
  }
}

__global__ void relu_inplace(float* __restrict__ x, size_t n) {
  size_t i = (size_t)blockIdx.x * blockDim.x + threadIdx.x;
  if (i < n) x[i] = fmaxf(x[i], 0.f);
}

// in-place log_softmax over 64 channels, one wave per node (2 ch / lane)
__global__ void log_softmax64(float* __restrict__ out, int nN) {
  int node = (int)((blockIdx.x * (size_t)blockDim.x + threadIdx.x) >> 5);
  int lane = threadIdx.x & 31;
  if (node >= nN) return;
  size_t base = (size_t)node * 64;
  float v0 = out[base + lane];
  float v1 = out[base + lane + 32];
  float m = fmaxf(v0, v1);
#pragma unroll
  for (int off = 16; off > 0; off >>= 1) m = fmaxf(m, __shfl_xor(m, off, 32));
  float sum = expf(v0 - m) + expf(v1 - m);
#pragma unroll
  for (int off = 16; off > 0; off >>= 1) sum += __shfl_xor(sum, off, 32);
  float ls = m + logf(sum);
  out[base + lane] = v0 - ls;
  out[base + lane + 32] = v1 - ls;
}

// ---------------------------------------------------------------------------
static inline size_t align256(size_t x) { return (x + 255) & ~(size_t)255; }

extern "C" void kernel_launch(void* const* d_in, const int* in_sizes, int n_in,
                              void* d_out, int out_size, void* d_ws,
                              size_t ws_size, hipStream_t stream) {
  constexpr int IN_C = 256, HID_C = 128, OUT_C = 64;
  const float* x = (const float*)d_in[0];
  const float* W1 = (const float*)d_in[1];
  const float* as1 = (const float*)d_in[2];
  const float* ad1 = (const float*)d_in[3];
  const float* b1 = (const float*)d_in[4];
  const float* W2 = (const float*)d_in[5];
  const float* as2 = (const float*)d_in[6];
  const float* ad2 = (const float*)d_in[7];
  const float* b2 = (const float*)d_in[8];
  const long long* ei = (const long long*)d_in[9];

  const int n = in_sizes[0] / IN_C;       // 100000
  const int E = in_sizes[9] / 2;          // 1600000
  const int eTot = E + n;                 // + self loops
  float* outp = (float*)d_out;

  // workspace layout
  char* ws = (char*)d_ws;
  size_t o = 0;
  float* h1 = (float*)(ws + o);           // n*HID, reused as h2 (n*OUT)
  o += align256((size_t)n * HID_C * 4);
  float* agg1 = (float*)(ws + o);         // n*HID, relu'd in place -> hid
  o += align256((size_t)n * HID_C * 4);
  float* a_s = (float*)(ws + o); o += align256((size_t)n * 4);
  float* a_d = (float*)(ws + o); o += align256((size_t)n * 4);
  unsigned* m_enc = (unsigned*)(ws + o); o += align256((size_t)n * 4);
  float* den = (float*)(ws + o); o += align256((size_t)n * 4);
  float* ebuf = (float*)(ws + o); o += align256((size_t)eTot * 4);
  (void)ws_size;
  float* h2 = h1;  // h1 dead after layer-1 aggregation

  const int gM = (n + 15) / 16;
  const int gNode = (n + 7) / 8;          // 8 waves/block, wave per node
  const int gEdge = (eTot + 255) / 256;
  const int gEdgeW = (eTot + 7) / 8;      // wave per edge

  // ----- layer 1 -----
  gat_gemm_wmma<IN_C, HID_C, 8><<<gM, 256, 0, stream>>>(x, W1, h1, n);
  node_prep<HID_C><<<gNode, 256, 0, stream>>>(h1, as1, ad1, a_s, a_d, m_enc,
                                              den, n);
  init_out<HID_C><<<(int)(((size_t)n * HID_C + 255) / 256), 256, 0, stream>>>(
      agg1, b1, n);
  edge_logits<<<gEdge, 256, 0, stream>>>(ei, a_s, a_d, ebuf, m_enc, E, n);
  edge_exp<<<gEdge, 256, 0, stream>>>(ei, ebuf, m_enc, den, E, n);
  edge_aggregate<HID_C><<<gEdgeW, 256, 0, stream>>>(ei, ebuf, den, h1, agg1, E,
                                                    n);
  relu_inplace<<<(int)(((size_t)n * HID_C + 255) / 256), 256, 0, stream>>>(
      agg1, (size_t)n * HID_C);

  // ----- layer 2 -----
  gat_gemm_wmma<HID_C, OUT_C, 4><<<gM, 128, 0, stream>>>(agg1, W2, h2, n);
  node_prep<OUT_C><<<gNode, 256, 0, stream>>>(h2, as2, ad2, a_s, a_d, m_enc,
                                              den, n);
  init_out<OUT_C><<<(int)(((size_t)n * OUT_C + 255) / 256), 256, 0, stream>>>(
      outp, b2, n);
  edge_logits<<<gEdge, 256, 0, stream>>>(ei, a_s, a_d, ebuf, m_enc, E, n);
  edge_exp<<<gEdge, 256, 0, stream>>>(ei, ebuf, m_enc, den, E, n);
  edge_aggregate<OUT_C><<<gEdgeW, 256, 0, stream>>>(ei, ebuf, den, h2, outp, E,
                                                    n);
  log_softmax64<<<gNode, 256, 0, stream>>>(outp, n);
}